// ModelNew_50388556316990
// MI455X (gfx1250) — compile-verified
//
#include <hip/hip_runtime.h>

// ---------------------------------------------------------------------------
// Problem: out[b,o] = bias[o]*2 + sum_{c,j} stats[b,c,j] * coeff[o,c,j]
//   stats: per (b,c) image: [full sum, row0 sum, col0 sum, corner]
//   coeff: per (o,c) from 3x3 weight: [sum9, -row0sum, -col0sum, w00] * scale
// Shapes: x (64,64,128,128) f32, weight (64,128,3,3) f32, bias (128) f32
// Output: (64,128,1,1) f32
// Roofline: dominated by streaming x once (268 MB -> ~11.5us @ 23.3 TB/s).
// ---------------------------------------------------------------------------

#define B_DIM   64
#define C_DIM   64
#define O_DIM   128
#define HW      128
#define IMG_ELEMS (HW * HW)          // 16384 floats per image
#define K_DIM   (C_DIM * 4)          // 256
#define MULT_F  2.0f

typedef __attribute__((ext_vector_type(2))) float v2f;
typedef __attribute__((ext_vector_type(4))) float v4f;
typedef __attribute__((ext_vector_type(8))) float v8f;

// ---------------------------------------------------------------------------
// Kernel 1: per-image statistics.  One block per (b,c) image.
// 256 threads * 16 float4 NT loads = 16384 floats. Bandwidth-bound: 268 MB.
// stats layout: stats[(b*64 + c)*4 + j]  == stats[b*256 + c*4 + j]
// ---------------------------------------------------------------------------
__global__ __launch_bounds__(256) void stats_kernel(const float* __restrict__ x,
                                                    float* __restrict__ stats) {
    const int bc = blockIdx.x;                 // 0..4095
    const int t  = threadIdx.x;                // 0..255
    const v4f* img = (const v4f*)(x + (size_t)bc * IMG_ELEMS);

    float full = 0.0f, top = 0.0f, left = 0.0f, corner = 0.0f;

    #pragma unroll
    for (int c = 0; c < 16; ++c) {
        // float4 id = c*256 + t; covers elements [id*4, id*4+3]
        v4f v = __builtin_nontemporal_load(&img[c * 256 + t]);
        full += (v.x + v.y) + (v.z + v.w);
        if ((t & 31) == 0) {                   // element (c*256+t)*4 % 128 == 0 -> col 0
            left += v.x;
        }
        if (c == 0 && t < 32) {                // elements < 128 -> row 0
            top += (v.x + v.y) + (v.z + v.w);
            if (t == 0) corner = v.x;
        }
    }

    __shared__ v4f red[256];
    v4f mine = {full, top, left, corner};
    red[t] = mine;
    __syncthreads();
    #pragma unroll
    for (int s = 128; s > 0; s >>= 1) {
        if (t < s) {
            red[t] = red[t] + red[t + s];
        }
        __syncthreads();
    }
    if (t == 0) {
        v4f r = red[0];
        float* o = stats + (size_t)bc * 4;
        o[0] = r.x; o[1] = r.y; o[2] = r.z; o[3] = r.w;
    }
}

// ---------------------------------------------------------------------------
// Kernel 2: coeff[o*256 + c*4 + j] from weight[c][o][k][l] (w is (i,o,3,3)).
// ---------------------------------------------------------------------------
__global__ __launch_bounds__(256) void coeff_kernel(const float* __restrict__ w,
                                                    float* __restrict__ coeff) {
    const int idx = blockIdx.x * 256 + threadIdx.x;   // idx = o*64 + c
    if (idx >= O_DIM * C_DIM) return;
    const int o = idx >> 6;
    const int c = idx & 63;
    const float* wp = w + (size_t)(c * O_DIM + o) * 9;
    const float w00 = wp[0], w01 = wp[1], w02 = wp[2];
    const float w10 = wp[3], w11 = wp[4], w12 = wp[5];
    const float w20 = wp[6], w21 = wp[7], w22 = wp[8];

    const float scale = MULT_F / (256.0f * 256.0f);
    const float full   = ((w00 + w01) + (w02 + w10)) + ((w11 + w12) + (w20 + w21)) + w22;
    const float top    = -(w00 + w01 + w02);
    const float left   = -(w00 + w10 + w20);
    const float corner = w00;

    float* out = coeff + (size_t)idx * 4;
    out[0] = full   * scale;
    out[1] = top    * scale;
    out[2] = left   * scale;
    out[3] = corner * scale;
}

// ---------------------------------------------------------------------------
// Kernel 3: 64x128 GEMM, K=256, via V_WMMA_F32_16X16X4_F32.
// One wave (32 threads) per 16x16 output tile; 4x8 tiles.
//   A = stats  (M=64 x K=256), row-major
//   B = coeff^T viewed as K x N: B[k][n] = coeff[n*256 + k]
//   D[m][n] += bias[n]*MULT
// Lane layout per ISA (32-bit A 16x4): lanes 0-15 hold K=0,1 in v[0],v[1];
// lanes 16-31 hold K=2,3.  C/D: vgpr v -> M = v + 8*(lane>=16), N = lane%16.
// ---------------------------------------------------------------------------
__global__ __launch_bounds__(32) void gemm_wmma_kernel(const float* __restrict__ stats,
                                                       const float* __restrict__ coeff,
                                                       const float* __restrict__ bias,
                                                       float* __restrict__ out) {
    const int n0   = blockIdx.x * 16;    // output-channel tile (0..7)*16
    const int m0   = blockIdx.y * 16;    // batch tile          (0..3)*16
    const int lane = threadIdx.x;        // 0..31 (wave32)
    const int half = lane >> 4;          // 0: K+{0,1}, 1: K+{2,3}
    const int lm   = lane & 15;          // M (for A) / N (for B, D)

    const float* arow = stats + (size_t)(m0 + lm) * K_DIM;
    const float* brow = coeff + (size_t)(n0 + lm) * K_DIM;

    v8f acc = {};
    #pragma unroll 4
    for (int k0 = 0; k0 < K_DIM; k0 += 4) {
        const int ka = k0 + 2 * half;
        v2f a, b;
        a.x = arow[ka];     a.y = arow[ka + 1];
        b.x = brow[ka];     b.y = brow[ka + 1];
        acc = __builtin_amdgcn_wmma_f32_16x16x4_f32(
            /*neg_a=*/false, a, /*neg_b=*/false, b,
            /*c_mod=*/(short)0, acc, /*reuse_a=*/false, /*reuse_b=*/false);
    }

    const float bb = bias[n0 + lm] * MULT_F;
    #pragma unroll
    for (int v = 0; v < 8; ++v) {
        const int m = v + 8 * half;
        out[(size_t)(m0 + m) * O_DIM + (n0 + lm)] = acc[v] + bb;
    }
}

// ---------------------------------------------------------------------------
// Launch
// ---------------------------------------------------------------------------
extern "C" void kernel_launch(void* const* d_in, const int* in_sizes, int n_in,
                              void* d_out, int out_size, void* d_ws, size_t ws_size,
                              hipStream_t stream) {
    const float* x    = (const float*)d_in[0];   // (64,64,128,128)
    const float* w    = (const float*)d_in[1];   // (64,128,3,3)
    const float* bias = (const float*)d_in[2];   // (128,)
    float* out = (float*)d_out;                  // (64,128,1,1) -> 8192 floats

    float* stats = (float*)d_ws;                 // 64*256 floats  (64 KB)
    float* coeff = stats + B_DIM * K_DIM;        // 128*256 floats (128 KB)

    stats_kernel<<<B_DIM * C_DIM, 256, 0, stream>>>(x, stats);
    coeff_kernel<<<(O_DIM * C_DIM + 255) / 256, 256, 0, stream>>>(w, coeff);
    gemm_wmma_kernel<<<dim3(O_DIM / 16, B_DIM / 16), 32, 0, stream>>>(stats, coeff, bias, out);
}